// MeshConv_31482110280358
// MI455X (gfx1250) — compile-verified
//
#include <hip/hip_runtime.h>

typedef float v2f __attribute__((ext_vector_type(2)));
typedef float v8f __attribute__((ext_vector_type(8)));

#define C_DIM 128
#define LDS_STRIDE 132   // 132 mod 64 words == 4 -> conflict-free ds_load_b64 pattern

// ---------------------------------------------------------------------------
// Kernel 1: COO scatter SpMM.  agg[r] += vals[e] * x[cols[e]]
// One wave per edge; lane owns 4 consecutive channels.  x (102MB) lives in
// the 192MB L2, so gathers + fp32 atomics are L2 traffic, not HBM traffic.
// ---------------------------------------------------------------------------
__global__ __launch_bounds__(256)
void spmm_scatter_kernel(const float* __restrict__ x,
                         const long long* __restrict__ rows,
                         const long long* __restrict__ cols,
                         const float* __restrict__ vals,
                         float* __restrict__ agg,
                         int nnz) {
    const int wave_in_blk = threadIdx.x >> 5;
    const int lane        = threadIdx.x & 31;
    const int e = blockIdx.x * (blockDim.x >> 5) + wave_in_blk;
    if (e >= nnz) return;

    const long long r = rows[e];
    const long long c = cols[e];
    const float     v = vals[e];

    const float4* xrow = (const float4*)(x + (size_t)c * C_DIM);
    float4 xv = xrow[lane];                       // coalesced 512B/wave

    float* dst = agg + (size_t)r * C_DIM + lane * 4;
    unsafeAtomicAdd(dst + 0, v * xv.x);
    unsafeAtomicAdd(dst + 1, v * xv.y);
    unsafeAtomicAdd(dst + 2, v * xv.z);
    unsafeAtomicAdd(dst + 3, v * xv.w);
}

// ---------------------------------------------------------------------------
// Kernel 2: out = agg @ W^T + b via V_WMMA_F32_16X16X4_F32.
// W is staged into LDS once per block with gfx1250 async global->LDS copies
// (ASYNCcnt), padded to a 132-float stride for bank-conflict-free b64 reads.
// Each wave computes a 16-row x 128-col tile; the K loop is software-
// pipelined (prefetch A from global + 8 B fragments from LDS one step ahead).
// ---------------------------------------------------------------------------
__global__ __launch_bounds__(256)
void gemm_wmma_kernel(const float* __restrict__ agg,
                      const float* __restrict__ W,
                      const float* __restrict__ bias,
                      float* __restrict__ out,
                      int nrows) {
    extern __shared__ float sW[];   // C_DIM * LDS_STRIDE floats (66KB)

    // ---- async stage W: 4096 chunks of 4 floats, 16 per thread ----
#pragma unroll
    for (int i = 0; i < 16; ++i) {
        const int chunk = i * 256 + threadIdx.x;      // 0..4095
        const int row   = chunk >> 5;                 // 32 chunks per row
        const int cw    = (chunk & 31) * 4;           // word offset in row
        const float* g  = W + row * C_DIM + cw;
        const unsigned loff =
            (unsigned)(uintptr_t)(&sW[row * LDS_STRIDE + cw]);  // LDS byte offset
        asm volatile("global_load_async_to_lds_b128 %0, %1, off"
                     :: "v"(loff), "v"(g)
                     : "memory");
    }
    asm volatile("s_wait_asynccnt 0x0" ::: "memory");
    __syncthreads();

    const int wave_in_blk = threadIdx.x >> 5;
    const int lane        = threadIdx.x & 31;
    const int tile = blockIdx.x * (blockDim.x >> 5) + wave_in_blk;
    const int row0 = tile * 16;
    if (row0 >= nrows) return;     // after the barrier: safe

    const int l15  = lane & 15;    // M row (A) / N col (B)
    const int half = lane >> 4;    // selects K pair
    const int kof  = half * 2;

    v8f acc[8] = {};

    const float* arow = agg + (size_t)(row0 + l15) * C_DIM + kof;

    // prologue: fragment 0
    v2f a = *(const v2f*)(arow);
    v2f bb[8];
#pragma unroll
    for (int t = 0; t < 8; ++t)
        bb[t] = *(const v2f*)&sW[(t * 16 + l15) * LDS_STRIDE + kof];

#pragma unroll
    for (int k = 0; k < C_DIM; k += 4) {
        v2f a_n = a;
        v2f bn[8];
        if (k + 4 < C_DIM) {
            a_n = *(const v2f*)(arow + k + 4);
#pragma unroll
            for (int t = 0; t < 8; ++t)
                bn[t] = *(const v2f*)&sW[(t * 16 + l15) * LDS_STRIDE + (k + 4) + kof];
        } else {
#pragma unroll
            for (int t = 0; t < 8; ++t) bn[t] = bb[t];
        }

#pragma unroll
        for (int t = 0; t < 8; ++t)
            acc[t] = __builtin_amdgcn_wmma_f32_16x16x4_f32(
                false, a, false, bb[t], (short)0, acc[t], false, false);

        a = a_n;
#pragma unroll
        for (int t = 0; t < 8; ++t) bb[t] = bn[t];
    }

    // C/D layout: vgpr v, lanes 0-15 -> M=v, N=lane; lanes 16-31 -> M=v+8
#pragma unroll
    for (int t = 0; t < 8; ++t) {
        const int   col = t * 16 + l15;
        const float bv  = bias[col];
#pragma unroll
        for (int v = 0; v < 8; ++v) {
            const int m = v + half * 8;
            out[(size_t)(row0 + m) * C_DIM + col] = acc[t][v] + bv;
        }
    }
}

// ---------------------------------------------------------------------------
// Host-side launch.
// ---------------------------------------------------------------------------
extern "C" void kernel_launch(void* const* d_in, const int* in_sizes, int n_in,
                              void* d_out, int out_size, void* d_ws, size_t ws_size,
                              hipStream_t stream) {
    const float*     x    = (const float*)d_in[0];
    const long long* rows = (const long long*)d_in[1];
    const long long* cols = (const long long*)d_in[2];
    const float*     vals = (const float*)d_in[3];
    const float*     W    = (const float*)d_in[4];
    const float*     b    = (const float*)d_in[5];
    float*           out  = (float*)d_out;
    float*           agg  = (float*)d_ws;

    const int n   = in_sizes[0] / C_DIM;   // N_VERTS
    const int nnz = in_sizes[3];           // NNZ

    hipMemsetAsync(agg, 0, (size_t)n * C_DIM * sizeof(float), stream);

    // SpMM scatter: 8 waves (8 edges) per 256-thread block.
    {
        const int edges_per_block = 256 / 32;
        const int grid = (nnz + edges_per_block - 1) / edges_per_block;
        spmm_scatter_kernel<<<grid, 256, 0, stream>>>(x, rows, cols, vals, agg, nnz);
    }

    // WMMA GEMM: 8 waves (8 row-tiles of 16) per 256-thread block.
    {
        const int tiles = n / 16;                       // N divisible by 16
        const int tiles_per_block = 256 / 32;
        const int grid = (tiles + tiles_per_block - 1) / tiles_per_block;
        const size_t lds_bytes = (size_t)C_DIM * LDS_STRIDE * sizeof(float);
        gemm_wmma_kernel<<<grid, 256, lds_bytes, stream>>>(agg, W, b, out, n);
    }
}